// Block_80865644249896
// MI455X (gfx1250) — compile-verified
//
#include <hip/hip_runtime.h>

// Problem dims
#define BSZ 32
#define KSZ 16
#define NSZ 64
#define EPSF 1e-3f

typedef __attribute__((ext_vector_type(2))) float v2f;
typedef __attribute__((ext_vector_type(8))) float v8f;
typedef __attribute__((ext_vector_type(4))) unsigned int u32x4;
typedef __attribute__((ext_vector_type(4))) int i32x4;
typedef __attribute__((ext_vector_type(8))) int i32x8;

#if __has_builtin(__builtin_amdgcn_tensor_load_to_lds) && \
    __has_builtin(__builtin_amdgcn_s_wait_tensorcnt)
#define USE_TDM 1
#else
#define USE_TDM 0
#endif

#if USE_TDM
// Issue a 1-D Tensor-Data-Mover transfer: nElem 4-byte words, contiguous,
// global -> LDS. D# layout per CDNA5 ISA (group0/group1).
__device__ __forceinline__ void tdm_load_1d(void* lds_dst, const void* gsrc,
                                            int nElem) {
  unsigned long long ga = (unsigned long long)(size_t)gsrc;
  unsigned la = (unsigned)(size_t)lds_dst;
  u32x4 g0;
  g0[0] = 1u;                                   // count=1 (valid user D#)
  g0[1] = la;                                   // lds_addr
  g0[2] = (unsigned)(ga & 0xFFFFFFFFu);         // global_addr[31:0]
  g0[3] = (unsigned)((ga >> 32) & 0x1FFFFFFu)   // global_addr[56:32]
          | (2u << 30);                         // type = 2 ("image")
  i32x8 g1;
  g1[0] = 2 << 16;                              // wg_mask=0, data_size=4B
  g1[1] = (nElem & 0xFFFF) << 16;               // tensor_dim0[15:0]
  g1[2] = ((unsigned)nElem >> 16) | (1 << 16);  // tensor_dim0[31:16], tdim1=1
  g1[3] = (nElem & 0xFFFF) << 16;               // tile_dim0
  g1[4] = 0;                                    // tile_dim1/2 unused (1-D)
  g1[5] = nElem;                                // tensor_dim0_stride[31:0]
  g1[6] = 0;
  g1[7] = 0;
  i32x4 z4 = {};
#if __clang_major__ >= 23
  i32x8 z8 = {};
  __builtin_amdgcn_tensor_load_to_lds(g0, g1, z4, z4, z8, 0);
#else
  __builtin_amdgcn_tensor_load_to_lds(g0, g1, z4, z4, 0);
#endif
}
#endif

// ---------- CDNA5 FP32 WMMA fragment helpers (wave32) ----------
// A (16x4, MxK): lanes 0-15 M=lane (K=k0,k0+1); lanes 16-31 M=lane-16 (K=k0+2,3)
__device__ __forceinline__ v2f fragA(const float* m, int row0, int k0, int ld) {
  int lane = threadIdx.x & 31;
  const float* p = m + (row0 + (lane & 15)) * ld + k0 + 2 * (lane >> 4);
  v2f a; a.x = p[0]; a.y = p[1]; return a;
}
// B (4x16, KxN) from row-major storage: strided pair
__device__ __forceinline__ v2f fragB(const float* m, int k0, int col0, int ld) {
  int lane = threadIdx.x & 31;
  const float* p = m + (k0 + 2 * (lane >> 4)) * ld + col0 + (lane & 15);
  v2f b; b.x = p[0]; b.y = p[ld]; return b;
}
// B (4x16) from transposed (column-major) storage: contiguous -> ds_load_b64
__device__ __forceinline__ v2f fragBT(const float* mT, int k0, int col0, int ld) {
  int lane = threadIdx.x & 31;
  const float* p = mT + (col0 + (lane & 15)) * ld + k0 + 2 * (lane >> 4);
  v2f b; b.x = p[0]; b.y = p[1]; return b;
}
// C/D (16x16) store, row-major: VGPR r -> row r + 8*hi, lane&15 -> col
__device__ __forceinline__ void storeCD(float* m, int row0, int col0, int ld,
                                        v8f acc) {
  int lane = threadIdx.x & 31;
  float* p = m + (row0 + 8 * (lane >> 4)) * ld + col0 + (lane & 15);
#pragma unroll
  for (int r = 0; r < 8; ++r) p[r * ld] = acc[r];
}
// C/D store transposed (column-major image): contiguous -> 2x ds_store_b128
__device__ __forceinline__ void storeCDt(float* m, int row0, int col0, int ld,
                                         v8f acc) {
  int lane = threadIdx.x & 31;
  float* p = m + (col0 + (lane & 15)) * ld + row0 + 8 * (lane >> 4);
  ((float4*)p)[0] = make_float4(acc[0], acc[1], acc[2], acc[3]);
  ((float4*)p)[1] = make_float4(acc[4], acc[5], acc[6], acc[7]);
}

__device__ __forceinline__ v8f wmma4(v2f a, v2f b, v8f c) {
  return __builtin_amdgcn_wmma_f32_16x16x4_f32(false, a, false, b, (short)0, c,
                                               false, false);
}

// One-wave 16x16x16 FP32 matmul; B operand given as a transposed LDS image
__device__ __forceinline__ v8f mm16t(const float* A, const float* Bt) {
  v8f acc = {};
#pragma unroll
  for (int k = 0; k < 16; k += 4)
    acc = wmma4(fragA(A, 0, k, 16), fragBT(Bt, k, 0, 16), acc);
  return acc;
}

// ---------- scalar parameters ----------
__global__ void k_scalars(const float* ap, const float* gwp, const float* gcp,
                          float* sc) {
  if (threadIdx.x == 0) {
    sc[0] = log1pf(expf(ap[0]));                   // alpha = softplus
    sc[1] = 0.3f + 5.0f * (tanhf(gwp[0]) + 1.0f);  // gamma_w
    sc[2] = 0.3f + 5.0f * (tanhf(gcp[0]) + 1.0f);  // gamma_c (= c_c*alpha)
  }
}

// ---------- lambda_min of a 16x16 SPD matrix, one wave ----------
__device__ float wave_lammin(const float* A, float* v) {
  int lane = threadIdx.x & 31;
  bool act = lane < 16;
  float rs = 0.f;
  if (act) {
#pragma unroll
    for (int j = 0; j < 16; ++j) rs += fabsf(A[lane * 16 + j]);
  }
#pragma unroll
  for (int off = 16; off >= 1; off >>= 1) rs = fmaxf(rs, __shfl_xor(rs, off));
  float s = rs;  // Gershgorin bound on lambda_max
  if (act) v[lane] = 1.0f;
  float mu = 0.f;
  for (int it = 0; it < 16; ++it) {
    float y = 0.f;
    if (act) {
#pragma unroll
      for (int j = 0; j < 16; ++j)
        y += ((lane == j ? s : 0.f) - A[lane * 16 + j]) * v[j];
    }
    float nn = act ? y * y : 0.f;
#pragma unroll
    for (int off = 16; off >= 1; off >>= 1) nn += __shfl_xor(nn, off);
    nn = sqrtf(nn) + 1e-30f;
    if (act) v[lane] = y / nn;
    mu = nn;  // ||(sI-A)v|| -> lambda_max(sI-A)
  }
  return s - mu;
}

// 1/lambda_min for every (b,n) block of C  (2048 matrices, 4 waves/block)
__global__ void k_lammin(const float* __restrict__ C, float* __restrict__ laminv) {
  __shared__ float As[4][256];
  __shared__ float vs[4][16];
  int w = threadIdx.x >> 5, lane = threadIdx.x & 31;
  int mat = blockIdx.x * 4 + w;
  const float* src = C + mat * 256;
  for (int idx = lane; idx < 256; idx += 32) As[w][idx] = src[idx];
  float lm = wave_lammin(As[w], vs[w]);
  if (lane == 0) laminv[mat] = 1.0f / fmaxf(lm, 1e-12f);
}

// c_w[b] = gamma_w / (rho_Rx[b] * max_n 1/lambda_min)
__global__ void k_cw(const float* laminv, const float* rho, const float* sc,
                     float* cw) {
  __shared__ float red[64];
  int b = blockIdx.x, t = threadIdx.x;
  red[t] = laminv[b * 64 + t];
  __syncthreads();
  for (int off = 32; off >= 1; off >>= 1) {
    if (t < off) red[t] = fmaxf(red[t], red[t + off]);
    __syncthreads();
  }
  if (t == 0) cw[b] = sc[1] / (rho[b] * red[0]);
}

// ---------- batched 16x16 SPD inverse: Newton-Schulz, one wave / matrix ------
// C SPD => iterate X stays symmetric => one LDS image serves row- and
// column-major reads; intermediates stored transposed (b128 stores, b64 loads).
__global__ void k_inv16(const float* __restrict__ Ain, float* __restrict__ Xout) {
  __shared__ float lds[8][3 * 256];
  int w = threadIdx.x >> 5, lane = threadIdx.x & 31;
  int mat = blockIdx.x * 8 + w;
  float* As = lds[w];
  float* Xs = As + 256;
  float* Ts = Xs + 256;
  const float* src = Ain + mat * 256;
#if USE_TDM
  if (lane == 0) {  // TDM is wave-level; EXEC ignored
    tdm_load_1d(As, src, 256);
    __builtin_amdgcn_s_wait_tensorcnt(0);
  }
#else
  for (int idx = lane; idx < 256; idx += 32) As[idx] = src[idx];
#endif
  float rs = 0.f, cs = 0.f;
  if (lane < 16) {
#pragma unroll
    for (int j = 0; j < 16; ++j) {
      rs += fabsf(As[lane * 16 + j]);
      cs += fabsf(As[j * 16 + lane]);
    }
  }
#pragma unroll
  for (int off = 16; off >= 1; off >>= 1) {
    rs = fmaxf(rs, __shfl_xor(rs, off));
    cs = fmaxf(cs, __shfl_xor(cs, off));
  }
  float scale = 1.0f / (rs * cs + 1e-30f);
  for (int idx = lane; idx < 256; idx += 32)
    Xs[idx] = As[idx] * scale;  // X0 = A^T/(n1*ninf) = A*scale (A symmetric)
  for (int it = 0; it < 10; ++it) {
    v8f t = mm16t(As, Xs);  // T = A*X (X symmetric: col-major image == X)
    v8f e;
#pragma unroll
    for (int r = 0; r < 8; ++r) {
      int row = 8 * (lane >> 4) + r, col = lane & 15;
      e[r] = (row == col ? 2.0f : 0.0f) - t[r];  // E = 2I - T
    }
    storeCDt(Ts, 0, 0, 16, e);   // store E^T (read back via fragBT)
    v8f xn = mm16t(Xs, Ts);      // X' = X*E
    storeCDt(Xs, 0, 0, 16, xn);  // X' symmetric: transposed store is X'
  }
  for (int idx = lane; idx < 256; idx += 32) Xout[mat * 256 + idx] = Xs[idx];
}

// ---------- batched 64x64 inverse: Newton-Schulz, one block / matrix ----------
__global__ void k_inv64(const float* __restrict__ Ain, float* __restrict__ Xout,
                        int iters) {
  __shared__ float As[4096], P0[4096], P1[4096];
  __shared__ float red[130];
  int tid = threadIdx.x, w = tid >> 5, lane = tid & 31;
  int tm = w >> 2, tn = w & 3;
  const float* src = Ain + (size_t)blockIdx.x * 4096;
#if USE_TDM
  if (tid == 0) {
    tdm_load_1d(As, src, 4096);
    __builtin_amdgcn_s_wait_tensorcnt(0);
  }
#else
  for (int idx = tid; idx < 1024; idx += 512)
    ((float4*)As)[idx] = ((const float4*)src)[idx];
#endif
  __syncthreads();
  if (tid < 64) {
    float s = 0;
    for (int j = 0; j < 64; ++j) s += fabsf(As[tid * 64 + j]);
    red[tid] = s;
  } else if (tid < 128) {
    int c = tid - 64;
    float s = 0;
    for (int j = 0; j < 64; ++j) s += fabsf(As[j * 64 + c]);
    red[tid] = s;
  }
  __syncthreads();
  if (tid == 0) {
    float ninf = 0, n1 = 0;
    for (int j = 0; j < 64; ++j) {
      ninf = fmaxf(ninf, red[j]);
      n1 = fmaxf(n1, red[64 + j]);
    }
    red[128] = 1.0f / (ninf * n1 + 1e-30f);
  }
  __syncthreads();
  float scale = red[128];
  for (int idx = tid; idx < 4096; idx += 512)
    P0[idx] = As[(idx & 63) * 64 + (idx >> 6)] * scale;  // X0 = A^T * scale
  __syncthreads();
  for (int it = 0; it < iters; ++it) {
    v8f t = {};
#pragma unroll
    for (int kk = 0; kk < 64; kk += 4)
      t = wmma4(fragA(As, 16 * tm, kk, 64), fragB(P0, kk, 16 * tn, 64), t);
    storeCDt(P1, 16 * tm, 16 * tn, 64, t);  // T = A*X stored transposed
    __syncthreads();
    v8f acc = {};
#pragma unroll
    for (int kk = 0; kk < 64; kk += 4)
      acc = wmma4(fragA(P0, 16 * tm, kk, 64), fragBT(P1, kk, 16 * tn, 64), acc);
    v8f xn;
#pragma unroll
    for (int r = 0; r < 8; ++r) {
      int row = 16 * tm + 8 * (lane >> 4) + r, col = 16 * tn + (lane & 15);
      xn[r] = 2.0f * P0[row * 64 + col] - acc[r];  // X' = 2X - X*T
    }
    __syncthreads();
    storeCD(P0, 16 * tm, 16 * tn, 64, xn);
    __syncthreads();
  }
  float* dst = Xout + (size_t)blockIdx.x * 4096;
  for (int idx = tid; idx < 4096; idx += 512) dst[idx] = P0[idx];
}

// ---------- quad + gradient step:  Wg = W - c_w*(quad - W^-T) ----------
// quad[b,k,n,i] = sum_l Cinv[b,n,k,l] * (Rx[b,k,l] * W[b,l]^T)[i,n]
__global__ void k_quad(const float* __restrict__ Rx, const float* __restrict__ Win,
                       const float* __restrict__ Winv, const float* __restrict__ Cinv,
                       const float* __restrict__ cw, float* __restrict__ Wg) {
  __shared__ float RxS[4096], Ws[4096];
  int tid = threadIdx.x, w = tid >> 5, lane = tid & 31;
  int tm = w >> 2, tn = w & 3;
  int b = blockIdx.x >> 4, k = blockIdx.x & 15;
  int n_lane = 16 * tn + (lane & 15);
  v8f q = {};
  for (int l = 0; l < 16; ++l) {
    const float* rsrc = Rx + (((size_t)(b * 16 + k) * 16 + l) << 12);
    const float* wsrc = Win + ((size_t)(b * 16 + l) << 12);
#if USE_TDM
    if (tid == 0) {  // DMA both 16KB tiles, then drain TENSORcnt
      tdm_load_1d(RxS, rsrc, 4096);
      tdm_load_1d(Ws, wsrc, 4096);
      __builtin_amdgcn_s_wait_tensorcnt(0);
    }
#else
    for (int idx = tid; idx < 1024; idx += 512) {
      ((float4*)RxS)[idx] = ((const float4*)rsrc)[idx];
      ((float4*)Ws)[idx] = ((const float4*)wsrc)[idx];
    }
#endif
    if (l < 15) __builtin_prefetch(rsrc + 4096 + tid * 8, 0, 1);  // warm L2
    __syncthreads();
    v8f m = {};
#pragma unroll
    for (int kk = 0; kk < 64; kk += 4)
      m = wmma4(fragA(RxS, 16 * tm, kk, 64), fragBT(Ws, kk, 16 * tn, 64), m);
    float cv = Cinv[(((b * 64) + n_lane) * 16 + k) * 16 + l];
#pragma unroll
    for (int r = 0; r < 8; ++r) q[r] += cv * m[r];
    __syncthreads();
  }
  float cwb = cw[b];
  size_t base = (size_t)(b * 16 + k) << 12;
#pragma unroll
  for (int r = 0; r < 8; ++r) {
    int ig = 16 * tm + 8 * (lane >> 4) + r;
    float wv = Win[base + n_lane * 64 + ig];
    float iv = Winv[base + ig * 64 + n_lane];  // W^-T[n][i] = Winv[i][n]
    Wg[base + n_lane * 64 + ig] = wv - cwb * (q[r] - iv);
  }
}

// prox_f(Wg) ~= Wg + c_w * Wg^-T   (1st order of s->0.5(s+sqrt(s^2+4c)))
__global__ void k_addT(const float* __restrict__ Wg, const float* __restrict__ WgInv,
                       const float* __restrict__ cw, float* __restrict__ Wout) {
  int idx = blockIdx.x * 256 + threadIdx.x;  // < 2097152
  int bk = idx >> 12;
  int b = bk >> 4;
  int n = (idx >> 6) & 63, i = idx & 63;
  Wout[idx] = Wg[idx] + cw[b] * WgInv[((size_t)bk << 12) + i * 64 + n];
}

// ---------- S[b,n,k,l] = w_k(n)^T Rx[k,l] w_l(n) ----------
__global__ void k_S(const float* __restrict__ Rx, const float* __restrict__ W,
                    float* __restrict__ S) {
  __shared__ float RxS[4096], Ws[4096], Ys[4096];
  int tid = threadIdx.x, w = tid >> 5;
  int tm = w >> 2, tn = w & 3;
  int l = blockIdx.x & 15, k = (blockIdx.x >> 4) & 15, b = blockIdx.x >> 8;
  const float* rsrc = Rx + ((size_t)blockIdx.x << 12);
  const float* wsrc = W + ((size_t)(b * 16 + l) << 12);
#if USE_TDM
  if (tid == 0) {
    tdm_load_1d(RxS, rsrc, 4096);
    tdm_load_1d(Ws, wsrc, 4096);
    __builtin_amdgcn_s_wait_tensorcnt(0);
  }
#else
  for (int idx = tid; idx < 1024; idx += 512) {
    ((float4*)RxS)[idx] = ((const float4*)rsrc)[idx];
    ((float4*)Ws)[idx] = ((const float4*)wsrc)[idx];
  }
#endif
  __syncthreads();
  v8f m = {};
#pragma unroll
  for (int kk = 0; kk < 64; kk += 4)
    m = wmma4(fragA(RxS, 16 * tm, kk, 64), fragBT(Ws, kk, 16 * tn, 64), m);
  storeCD(Ys, 16 * tm, 16 * tn, 64, m);  // Y = Rx * W_l^T  (i,n)
  __syncthreads();
  if (tid < 64) {
    int n = tid;
    const float* wk = W + ((size_t)(b * 16 + k) << 12) + n * 64;
    float s = 0.f;
    for (int i = 0; i < 64; ++i) s += wk[i] * Ys[i * 64 + n];
    S[(((b * 64) + n) * 16 + k) * 16 + l] = s;
  }
}

// ---------- C gradient step + symmetrize ----------
__global__ void k_cstep(const float* __restrict__ Cin, const float* __restrict__ S,
                        const float* __restrict__ sc, float* __restrict__ Cs) {
  int idx = blockIdx.x * 256 + threadIdx.x;  // < 524288
  int base = idx & ~255;
  int kk = (idx >> 4) & 15, ll = idx & 15;
  int tix = base + ll * 16 + kk;
  float t = sc[2];  // c_c * alpha == gamma_c
  float g1 = Cin[idx] - t * (Cin[idx] - S[idx]);
  float g2 = Cin[tix] - t * (Cin[tix] - S[tix]);
  Cs[idx] = 0.5f * (g1 + g2);
}

// PSD floor: shift spectrum so lambda_min >= eps
__global__ void k_cproj(const float* __restrict__ Cin, float* __restrict__ Cout) {
  __shared__ float As[4][256];
  __shared__ float vs[4][16];
  int w = threadIdx.x >> 5, lane = threadIdx.x & 31;
  int mat = blockIdx.x * 4 + w;
  const float* src = Cin + mat * 256;
  for (int idx = lane; idx < 256; idx += 32) As[w][idx] = src[idx];
  float lm = wave_lammin(As[w], vs[w]);
  float shift = fmaxf(0.0f, EPSF - lm);
  for (int idx = lane; idx < 256; idx += 32)
    Cout[mat * 256 + idx] =
        As[w][idx] + (((idx >> 4) == (idx & 15)) ? shift : 0.0f);
}

extern "C" void kernel_launch(void* const* d_in, const int* in_sizes, int n_in,
                              void* d_out, int out_size, void* d_ws, size_t ws_size,
                              hipStream_t stream) {
  const float* Rx  = (const float*)d_in[0];
  const float* rho = (const float*)d_in[1];
  const float* W0  = (const float*)d_in[2];
  const float* C0  = (const float*)d_in[3];
  const float* ap  = (const float*)d_in[4];
  const float* gwp = (const float*)d_in[5];
  const float* gcp = (const float*)d_in[6];

  float* out = (float*)d_out;
  float* Wfin = out;            // (B,K,N,N) = 2097152
  float* Cfin = out + 2097152;  // (B,N,K,K) = 524288

  float* ws = (float*)d_ws;
  float* sc     = ws;              // 8
  float* cw     = ws + 8;          // 32
  float* laminv = ws + 64;         // 2048
  float* Cinv   = ws + 2112;       // 524288
  float* Winv   = Cinv + 524288;   // 2097152
  float* Wg     = Winv + 2097152;  // 2097152
  float* WA     = Wg + 2097152;    // 2097152
  float* WB     = WA + 2097152;    // 2097152
  float* Sbuf   = WB + 2097152;    // 524288
  float* Csym   = Sbuf + 524288;   // 524288
  float* CA     = Csym + 524288;   // 524288
  float* CB     = CA + 524288;     // 524288

  k_scalars<<<1, 32, 0, stream>>>(ap, gwp, gcp, sc);
  k_lammin<<<512, 128, 0, stream>>>(C0, laminv);
  k_cw<<<32, 64, 0, stream>>>(laminv, rho, sc, cw);
  k_inv16<<<256, 256, 0, stream>>>(C0, Cinv);

  const float* Wi = W0;
  float* Wo[3] = {WA, WB, Wfin};
  for (int it = 0; it < 3; ++it) {
    k_inv64<<<512, 512, 0, stream>>>(Wi, Winv, 12);
    k_quad<<<512, 512, 0, stream>>>(Rx, Wi, Winv, Cinv, cw, Wg);
    k_inv64<<<512, 512, 0, stream>>>(Wg, Winv, 12);
    k_addT<<<8192, 256, 0, stream>>>(Wg, Winv, cw, Wo[it]);
    Wi = Wo[it];
  }

  k_S<<<8192, 512, 0, stream>>>(Rx, Wfin, Sbuf);

  const float* Ci = C0;
  float* Co[3] = {CA, CB, Cfin};
  for (int it = 0; it < 3; ++it) {
    k_cstep<<<2048, 256, 0, stream>>>(Ci, Sbuf, sc, Csym);
    k_cproj<<<512, 128, 0, stream>>>(Csym, Co[it]);
    Ci = Co[it];
  }
}